// ResGEChebNet_35562329211079
// MI455X (gfx1250) — compile-verified
//
#include <hip/hip_runtime.h>
#include <hip/hip_bf16.h>

typedef float v2f __attribute__((ext_vector_type(2)));
typedef float v8f __attribute__((ext_vector_type(8)));
typedef int   v4i __attribute__((ext_vector_type(4)));

#define GV 50000
#define GE 800000
#define GB 8
#define BVROWS (GB * GV)            /* 400000 */
#define MTILES (BVROWS / 16)        /* 25000  */
#define GEMM_WAVES 8
#define GEMM_BLOCKS (MTILES / GEMM_WAVES) /* 3125 */

#define GEMM_INIT 1
#define GEMM_RELU 2

#define AS1 __attribute__((address_space(1)))
#define AS3 __attribute__((address_space(3)))

#if __has_builtin(__builtin_amdgcn_global_load_async_to_lds_b128) && \
    __has_builtin(__builtin_amdgcn_s_wait_asynccnt)
#define USE_ASYNC_LDS 1
#endif

// ---------------------------------------------------------------------------
// Elementwise kernels
// ---------------------------------------------------------------------------
__global__ void k_transpose_in(const float* __restrict__ x, float* __restrict__ act, int C) {
  // x: (B, C, V) -> act: (B, V, C)
  size_t t = (size_t)blockIdx.x * blockDim.x + threadIdx.x;
  size_t total = (size_t)GB * C * GV;
  if (t >= total) return;
  int c = (int)(t % C);
  size_t r = t / C;
  int v = (int)(r % GV);
  int b = (int)(r / GV);
  act[t] = x[((size_t)b * C + c) * GV + v];
}

__global__ void k_zero(float* __restrict__ p, size_t n) {
  size_t t = (size_t)blockIdx.x * blockDim.x + threadIdx.x;
  if (t < n) p[t] = 0.0f;
}

__global__ void k_scale_copy(float* __restrict__ dst, const float* __restrict__ src,
                             float a, size_t n) {
  size_t t = (size_t)blockIdx.x * blockDim.x + threadIdx.x;
  if (t < n) dst[t] = a * src[t];
}

__global__ void k_scale_inplace(float* __restrict__ p, float a, size_t n) {
  size_t t = (size_t)blockIdx.x * blockDim.x + threadIdx.x;
  if (t < n) p[t] *= a;
}

__global__ void k_add_relu(float* __restrict__ dst, const float* __restrict__ a,
                           const float* __restrict__ b, size_t n) {
  size_t t = (size_t)blockIdx.x * blockDim.x + threadIdx.x;
  if (t < n) dst[t] = fmaxf(a[t] + b[t], 0.0f);
}

// ---------------------------------------------------------------------------
// SpMM: dst[(b*V+row)*C + c] += scale * vals[e] * src[(b*V+col)*C + c]
// Group of B*C consecutive threads handles one edge; lanes map to (b, c) so
// gathers/scatters are contiguous C-float rows -> coalesced. Activation tensor
// (<=102MB) fits in the 192MB L2, so random per-edge row gathers hit L2.
// ---------------------------------------------------------------------------
__global__ void k_spmm(float* __restrict__ dst, const float* __restrict__ src,
                       const int* __restrict__ rows, const int* __restrict__ cols,
                       const float* __restrict__ vals, int C, float scale) {
  size_t t = (size_t)blockIdx.x * blockDim.x + threadIdx.x;
  int lanesPerEdge = GB * C;
  size_t e = t / lanesPerEdge;
  if (e >= (size_t)GE) return;
  int g = (int)(t % lanesPerEdge);
  int b = g / C;
  int c = g % C;
  int row = rows[e];
  int col = cols[e];
  float w = vals[e] * scale;
  const float* sp = src + ((size_t)b * GV + col) * C + c;
  float* dp = dst + ((size_t)b * GV + row) * C + c;
  __builtin_prefetch(dp, 1, 3);  // global_prefetch_b8 on the scatter target
  atomicAdd(dp, w * sp[0]);
}

// ---------------------------------------------------------------------------
// GEMM with f32 WMMA, compile-time (CIN, COUT):
//   flags&GEMM_INIT: out = A@W + bias   (no read of out)
//   else:            out += A@W
//   flags&GEMM_RELU: fused ReLU on the final value.
// One wave computes a 16-row M tile and COUT/16 N tiles, reusing the A
// fragment. W slice staged in LDS via async-to-LDS DMA (ASYNCcnt) when the
// toolchain exposes the gfx1250 builtins; constant-offset ds_load in the fully
// unrolled K loop.
// V_WMMA_F32_16X16X4_F32 fragment layouts:
//   A 16x4:  lane = half*16 + m ; vgpr0 = A[m][2*half], vgpr1 = A[m][2*half+1]
//   B 4x16:  lane = half*16 + n ; vgpr0 = B[2*half][n], vgpr1 = B[2*half+1][n]
//   C 16x16: vgpr r: lanes0-15 -> (M=r, N=lane); lanes16-31 -> (M=r+8, N=lane-16)
// ---------------------------------------------------------------------------
template <int CIN, int COUT>
__global__ void k_gemm(float* __restrict__ out, const float* __restrict__ A,
                       const float* __restrict__ W, const float* __restrict__ bias,
                       int flags) {
  constexpr int NT = COUT / 16;
  __shared__ float Wlds[CIN * COUT];
  int tid = threadIdx.x;
#ifdef USE_ASYNC_LDS
#pragma unroll
  for (int i = tid * 4; i < CIN * COUT; i += 256 * 4) {
    __builtin_amdgcn_global_load_async_to_lds_b128(
        (AS1 v4i*)(W + i), (AS3 v4i*)(Wlds + i), 0, 0);
  }
  __builtin_amdgcn_s_wait_asynccnt(0);
  __syncthreads();
#else
#pragma unroll
  for (int i = tid; i < CIN * COUT; i += 256) Wlds[i] = W[i];
  __syncthreads();
#endif

  int wave = tid >> 5;
  int lane = tid & 31;
  size_t mtile = (size_t)blockIdx.x * GEMM_WAVES + wave;
  size_t rowbase = mtile * 16;

  int half = lane >> 4;       // 0 or 1
  int l15  = lane & 15;
  int klo  = half * 2;

  const v8f vzero = {0.f, 0.f, 0.f, 0.f, 0.f, 0.f, 0.f, 0.f};
  v8f acc[NT];
#pragma unroll
  for (int nt = 0; nt < NT; ++nt) acc[nt] = vzero;

  const float* Arow = A + (rowbase + (size_t)l15) * CIN + klo;
  const float* Brow = Wlds + klo * COUT + l15;

#pragma unroll
  for (int kk = 0; kk < CIN; kk += 4) {
    // 8-byte aligned pair load of A fragment
    v2f a = *reinterpret_cast<const v2f*>(Arow + kk);
#pragma unroll
    for (int nt = 0; nt < NT; ++nt) {
      v2f bf;
      bf.x = Brow[kk * COUT + nt * 16];
      bf.y = Brow[(kk + 1) * COUT + nt * 16];
      acc[nt] = __builtin_amdgcn_wmma_f32_16x16x4_f32(
          false, a, false, bf, (short)0, acc[nt], false, false);
    }
  }

  const int initOut = flags & GEMM_INIT;
  const int relu    = flags & GEMM_RELU;
  float* Orow = out + (rowbase + (size_t)half * 8) * COUT + l15;
#pragma unroll
  for (int nt = 0; nt < NT; ++nt) {
    float bv = (initOut && bias) ? bias[nt * 16 + l15] : 0.0f;
#pragma unroll
    for (int r = 0; r < 8; ++r) {
      size_t idx = (size_t)r * COUT + nt * 16;
      float base = initOut ? bv : Orow[idx];
      float v = base + acc[nt][r];
      if (relu) v = fmaxf(v, 0.0f);
      Orow[idx] = v;
    }
  }
}

// ---------------------------------------------------------------------------
// BatchNorm: stats (sum, sumsq per channel) then normalize.
// ---------------------------------------------------------------------------
__global__ void k_bn_stats(const float* __restrict__ x, float* __restrict__ stats, int C) {
  __shared__ float ssum[64];
  __shared__ float ssq[64];
  int tid = threadIdx.x;
  for (int i = tid; i < C; i += blockDim.x) { ssum[i] = 0.f; ssq[i] = 0.f; }
  __syncthreads();
  size_t total = (size_t)GB * GV * C;
  size_t stride = (size_t)gridDim.x * blockDim.x;
  for (size_t t = (size_t)blockIdx.x * blockDim.x + tid; t < total; t += stride) {
    float v = x[t];
    int c = (int)(t % C);
    atomicAdd(&ssum[c], v);
    atomicAdd(&ssq[c], v * v);
  }
  __syncthreads();
  for (int i = tid; i < C; i += blockDim.x) {
    atomicAdd(&stats[i], ssum[i]);
    atomicAdd(&stats[64 + i], ssq[i]);
  }
}

__global__ void k_bn_apply(float* __restrict__ y, const float* __restrict__ x,
                           const float* __restrict__ g, const float* __restrict__ be,
                           const float* __restrict__ stats, int C) {
  size_t t = (size_t)blockIdx.x * blockDim.x + threadIdx.x;
  size_t total = (size_t)GB * GV * C;
  if (t >= total) return;
  int c = (int)(t % C);
  const float n = (float)((size_t)GB * GV);
  float m = stats[c] / n;
  float var = stats[64 + c] / n - m * m;
  y[t] = (x[t] - m) * rsqrtf(var + 1e-5f) * g[c] + be[c];
}

// ---------------------------------------------------------------------------
// Head: pad (4,64,10) weights to (4,64,16), pooled max + log_softmax.
// ---------------------------------------------------------------------------
__global__ void k_pad_w(const float* __restrict__ w, float* __restrict__ wp) {
  int t = blockIdx.x * blockDim.x + threadIdx.x;
  const int total = 4 * 64 * 16;
  if (t >= total) return;
  int o = t % 16;
  int i = (t / 16) % 64;
  int k = t / (16 * 64);
  wp[t] = (o < 10) ? w[(k * 64 + i) * 10 + o] : 0.0f;
}

__global__ void k_pad_b(const float* __restrict__ b, float* __restrict__ bp) {
  int t = threadIdx.x;
  if (t < 16) bp[t] = (t < 10) ? b[t] : 0.0f;
}

__global__ void k_maxpool(const float* __restrict__ h, float* __restrict__ pooled) {
  // h: (B, V, 16) already ReLU'd; pooled: (B, 10); block per (b, o)
  int b = blockIdx.x / 10;
  int o = blockIdx.x % 10;
  __shared__ float red[256];
  float m = -3.0e38f;
  for (int v = threadIdx.x; v < GV; v += blockDim.x)
    m = fmaxf(m, h[((size_t)b * GV + v) * 16 + o]);
  red[threadIdx.x] = m;
  __syncthreads();
  for (int s = 128; s > 0; s >>= 1) {
    if ((int)threadIdx.x < s) red[threadIdx.x] = fmaxf(red[threadIdx.x], red[threadIdx.x + s]);
    __syncthreads();
  }
  if (threadIdx.x == 0) pooled[blockIdx.x] = red[0];
}

__global__ void k_logsoftmax(const float* __restrict__ pooled, float* __restrict__ out) {
  int b = threadIdx.x;
  if (b >= GB) return;
  float m = -3.0e38f;
  for (int o = 0; o < 10; ++o) m = fmaxf(m, pooled[b * 10 + o]);
  float s = 0.0f;
  for (int o = 0; o < 10; ++o) s += expf(pooled[b * 10 + o] - m);
  float ls = logf(s);
  for (int o = 0; o < 10; ++o) out[b * 10 + o] = pooled[b * 10 + o] - m - ls;
}

// ---------------------------------------------------------------------------
// Host-side helpers
// ---------------------------------------------------------------------------
static inline unsigned cdiv(size_t n, unsigned b) { return (unsigned)((n + b - 1) / b); }

static void run_gemm(float* out, const float* A, const float* W, const float* bias,
                     int Cin, int Cout, int flags, hipStream_t s) {
  if (Cin == 16 && Cout == 32)
    k_gemm<16, 32><<<GEMM_BLOCKS, 256, 0, s>>>(out, A, W, bias, flags);
  else if (Cin == 32 && Cout == 32)
    k_gemm<32, 32><<<GEMM_BLOCKS, 256, 0, s>>>(out, A, W, bias, flags);
  else if (Cin == 32 && Cout == 64)
    k_gemm<32, 64><<<GEMM_BLOCKS, 256, 0, s>>>(out, A, W, bias, flags);
  else if (Cin == 64 && Cout == 64)
    k_gemm<64, 64><<<GEMM_BLOCKS, 256, 0, s>>>(out, A, W, bias, flags);
  else
    k_gemm<64, 16><<<GEMM_BLOCKS, 256, 0, s>>>(out, A, W, bias, flags);
}

static void run_spmm(float* dst, const float* src, const int* rows, const int* cols,
                     const float* vals, int C, float scale, hipStream_t s) {
  size_t threads = (size_t)GE * GB * C;
  k_spmm<<<cdiv(threads, 256), 256, 0, s>>>(dst, src, rows, cols, vals, C, scale);
}

// Chebyshev graph conv; k processed sequentially. First GEMM initializes out
// with the bias, later GEMMs accumulate; optional ReLU fused into last GEMM.
static void chebconv(const float* in, int Cin, const float* W, const float* bias,
                     float* out, int Cout, int Kcheb, int reluLast,
                     float* Ta, float* Tb,
                     const int* rows, const int* cols, const float* vals,
                     hipStream_t s) {
  size_t nIn = (size_t)GB * GV * Cin;
  run_gemm(out, in, W, bias, Cin, Cout,
           GEMM_INIT | ((Kcheb == 1 && reluLast) ? GEMM_RELU : 0), s);
  if (Kcheb > 1) {
    // T1 = L x
    k_zero<<<cdiv(nIn, 256), 256, 0, s>>>(Ta, nIn);
    run_spmm(Ta, in, rows, cols, vals, Cin, 1.0f, s);
    run_gemm(out, Ta, W + 1 * Cin * Cout, nullptr, Cin, Cout,
             (Kcheb == 2 && reluLast) ? GEMM_RELU : 0, s);
    if (Kcheb > 2) {
      // T2 = 2 L T1 - T0
      k_scale_copy<<<cdiv(nIn, 256), 256, 0, s>>>(Tb, in, -1.0f, nIn);
      run_spmm(Tb, Ta, rows, cols, vals, Cin, 2.0f, s);
      run_gemm(out, Tb, W + 2 * Cin * Cout, nullptr, Cin, Cout,
               (Kcheb == 3 && reluLast) ? GEMM_RELU : 0, s);
      if (Kcheb > 3) {
        // T3 = 2 L T2 - T1 (in place over T1's buffer)
        k_scale_inplace<<<cdiv(nIn, 256), 256, 0, s>>>(Ta, -1.0f, nIn);
        run_spmm(Ta, Tb, rows, cols, vals, Cin, 2.0f, s);
        run_gemm(out, Ta, W + 3 * Cin * Cout, nullptr, Cin, Cout,
                 reluLast ? GEMM_RELU : 0, s);
      }
    }
  }
}

static void run_bn(float* y, const float* x, const float* g, const float* be, int C,
                   float* stats, hipStream_t s) {
  k_zero<<<1, 128, 0, s>>>(stats, 128);
  k_bn_stats<<<1024, 256, 0, s>>>(x, stats, C);
  size_t n = (size_t)GB * GV * C;
  k_bn_apply<<<cdiv(n, 256), 256, 0, s>>>(y, x, g, be, stats, C);
}

// ---------------------------------------------------------------------------
extern "C" void kernel_launch(void* const* d_in, const int* in_sizes, int n_in,
                              void* d_out, int out_size, void* d_ws, size_t ws_size,
                              hipStream_t stream) {
  const float* x     = (const float*)d_in[0];
  const int*   rows  = (const int*)d_in[1];
  const int*   cols  = (const int*)d_in[2];
  const float* vals  = (const float*)d_in[3];
  const float* W_in  = (const float*)d_in[4];
  const float* b_in  = (const float*)d_in[5];
  const float* g1a = (const float*)d_in[6],  *be1a = (const float*)d_in[7];
  const float* W1a = (const float*)d_in[8],  *b1a  = (const float*)d_in[9];
  const float* g1b = (const float*)d_in[10], *be1b = (const float*)d_in[11];
  const float* W1b = (const float*)d_in[12], *b1b  = (const float*)d_in[13];
  const float* g2a = (const float*)d_in[14], *be2a = (const float*)d_in[15];
  const float* W2a = (const float*)d_in[16], *b2a  = (const float*)d_in[17];
  const float* g2b = (const float*)d_in[18], *be2b = (const float*)d_in[19];
  const float* W2b = (const float*)d_in[20], *b2b  = (const float*)d_in[21];
  const float* W2s = (const float*)d_in[22];
  const float* g3a = (const float*)d_in[23], *be3a = (const float*)d_in[24];
  const float* W3a = (const float*)d_in[25], *b3a  = (const float*)d_in[26];
  const float* g3b = (const float*)d_in[27], *be3b = (const float*)d_in[28];
  const float* W3b = (const float*)d_in[29], *b3b  = (const float*)d_in[30];
  const float* g_out  = (const float*)d_in[31], *be_out = (const float*)d_in[32];
  const float* W_out  = (const float*)d_in[33], *b_out  = (const float*)d_in[34];

  float* ws = (float*)d_ws;
  const size_t NBUF = (size_t)GB * GV * 64;   // 25.6M floats per 64-ch buffer
  float* ACT   = ws + 0 * NBUF;               // block activation (node-major)
  float* XN    = ws + 1 * NBUF;               // bn output / shortcut
  float* MID   = ws + 2 * NBUF;               // mid conv output
  float* OUT2  = ws + 3 * NBUF;               // second conv output
  float* TA    = ws + 4 * NBUF;               // Chebyshev scratch
  float* TB    = ws + 5 * NBUF;               // Chebyshev scratch
  float* STATS = ws + 6 * NBUF;               // 128 floats (sum | sumsq)
  float* WPAD  = STATS + 128;                 // 4*64*16
  float* BPAD  = WPAD + 4096;                 // 16
  float* POOL  = BPAD + 16;                   // 80
  float* ACT0  = OUT2;                        // (B,V,16) transposed input staging

  // --- input transpose (B,C,V) -> (B,V,C), C=16 ---
  {
    size_t n = (size_t)GB * 16 * GV;
    k_transpose_in<<<cdiv(n, 256), 256, 0, stream>>>(x, ACT0, 16);
  }

  // --- conv_in: 16 -> 32, ReLU ---
  chebconv(ACT0, 16, W_in, b_in, ACT, 32, 4, 1, TA, TB, rows, cols, vals, stream);

  // --- block 1: 32 -> 32, identity shortcut ---
  run_bn(XN, ACT, g1a, be1a, 32, STATS, stream);
  chebconv(XN, 32, W1a, b1a, MID, 32, 4, 1, TA, TB, rows, cols, vals, stream);
  run_bn(MID, MID, g1b, be1b, 32, STATS, stream);
  chebconv(MID, 32, W1b, b1b, OUT2, 32, 4, 0, TA, TB, rows, cols, vals, stream);
  k_add_relu<<<cdiv((size_t)GB * GV * 32, 256), 256, 0, stream>>>(
      ACT, OUT2, XN, (size_t)GB * GV * 32);

  // --- block 2: 32 -> 64, K=1 conv shortcut ---
  run_bn(XN, ACT, g2a, be2a, 32, STATS, stream);
  chebconv(XN, 32, W2a, b2a, MID, 64, 4, 1, TA, TB, rows, cols, vals, stream);
  run_bn(MID, MID, g2b, be2b, 64, STATS, stream);
  chebconv(MID, 64, W2b, b2b, OUT2, 64, 4, 0, TA, TB, rows, cols, vals, stream);
  chebconv(XN, 32, W2s, nullptr, TA, 64, 1, 0, nullptr, nullptr, rows, cols, vals, stream);
  k_add_relu<<<cdiv((size_t)GB * GV * 64, 256), 256, 0, stream>>>(
      ACT, OUT2, TA, (size_t)GB * GV * 64);

  // --- block 3: 64 -> 64, identity shortcut ---
  run_bn(XN, ACT, g3a, be3a, 64, STATS, stream);
  chebconv(XN, 64, W3a, b3a, MID, 64, 4, 1, TA, TB, rows, cols, vals, stream);
  run_bn(MID, MID, g3b, be3b, 64, STATS, stream);
  chebconv(MID, 64, W3b, b3b, OUT2, 64, 4, 0, TA, TB, rows, cols, vals, stream);
  k_add_relu<<<cdiv((size_t)GB * GV * 64, 256), 256, 0, stream>>>(
      ACT, OUT2, XN, (size_t)GB * GV * 64);

  // --- head: bn, conv 64 -> 10 (padded to 16), ReLU, maxpool, log_softmax ---
  run_bn(XN, ACT, g_out, be_out, 64, STATS, stream);
  k_pad_w<<<cdiv(4 * 64 * 16, 256), 256, 0, stream>>>(W_out, WPAD);
  k_pad_b<<<1, 32, 0, stream>>>(b_out, BPAD);
  chebconv(XN, 64, WPAD, BPAD, MID, 16, 4, 1, TA, TB, rows, cols, vals, stream);
  k_maxpool<<<GB * 10, 256, 0, stream>>>(MID, POOL);
  k_logsoftmax<<<1, 32, 0, stream>>>(POOL, (float*)d_out);
}